// TestWrapperModule_7232724927052
// MI455X (gfx1250) — compile-verified
//
#include <hip/hip_runtime.h>

typedef __attribute__((ext_vector_type(2))) float v2f;
typedef __attribute__((ext_vector_type(8))) float v8f;

#define N_ROWS  65536
#define K_DIM   256
#define OUT_DIM 256
#define NNZ_CNT 2048

// ---------------------------------------------------------------------------
// Step 1: densify the sparse CG matrix into column-major W  (Wt[j*256 + i])
// so that WMMA B-fragments are contiguous 8-byte per-lane loads.
// ---------------------------------------------------------------------------
__global__ void zero_w_kernel(float* __restrict__ Wt) {
    Wt[blockIdx.x * blockDim.x + threadIdx.x] = 0.0f;
}

__global__ void scatter_w_kernel(const float* __restrict__ scale,
                                 const int*   __restrict__ M_in,
                                 const int*   __restrict__ M,
                                 float*       __restrict__ Wt) {
    int k = blockIdx.x * blockDim.x + threadIdx.x;
    if (k < NNZ_CNT) {
        // W[M_in[k]][M[k]] += scale[k]  stored transposed: Wt[M[k]][M_in[k]]
        atomicAdd(&Wt[(size_t)M[k] * K_DIM + M_in[k]], scale[k]);
    }
}

// ---------------------------------------------------------------------------
// Step 2: out[65536x256] = input[65536x256] * W[256x256], full f32 via
// V_WMMA_F32_16X16X4_F32.  One wave -> 16 rows x 64 cols (4 N-tiles),
// A-fragment reused across the 4 WMMAs each K-step.
//
// 16x4 f32 A layout (ISA 7.12.2): lanes 0-15 hold {K=kb,kb+1} of row M=lane,
// lanes 16-31 hold {K=kb+2,kb+3} -> per-lane v2f index = kb/2 + half.
// B (4x16) mirrors with N across lanes; Wt row-contiguity gives b64 loads.
// C/D: lane l, VGPR r -> C[(l>>4)*8 + r][l&15].
// ---------------------------------------------------------------------------
__global__ __launch_bounds__(256) void gemm_wmma_kernel(const float* __restrict__ A,
                                                        const float* __restrict__ Wt,
                                                        float*       __restrict__ C) {
    const int lane = threadIdx.x & 31;
    const int wave = threadIdx.x >> 5;
    const int half = lane >> 4;     // 0: lanes 0-15, 1: lanes 16-31
    const int lid  = lane & 15;

    const int gwave   = blockIdx.x * 8 + wave;
    const int rowTile = gwave >> 2;       // 0..4095  (16 rows each)
    const int colGrp  = gwave & 3;        // 0..3     (64 cols each)
    const int i0  = rowTile * 16;
    const int jb0 = colGrp * 64;

    const v2f* __restrict__ arow = reinterpret_cast<const v2f*>(A  + (size_t)(i0  + lid) * K_DIM);
    const v2f* __restrict__ b0   = reinterpret_cast<const v2f*>(Wt + (size_t)(jb0 + lid) * K_DIM);
    const v2f* __restrict__ b1   = b0 + (16 * K_DIM) / 2;
    const v2f* __restrict__ b2   = b0 + (32 * K_DIM) / 2;
    const v2f* __restrict__ b3   = b0 + (48 * K_DIM) / 2;

    v8f c0 = {}, c1 = {}, c2 = {}, c3 = {};

    #pragma unroll 4
    for (int kb = 0; kb < K_DIM; kb += 4) {
        const int fi = (kb >> 1) + half;
        const v2f a  = arow[fi];
        const v2f bA = b0[fi];
        const v2f bB = b1[fi];
        const v2f bC = b2[fi];
        const v2f bD = b3[fi];
        c0 = __builtin_amdgcn_wmma_f32_16x16x4_f32(false, a, false, bA, (short)0, c0, false, false);
        c1 = __builtin_amdgcn_wmma_f32_16x16x4_f32(false, a, false, bB, (short)0, c1, false, false);
        c2 = __builtin_amdgcn_wmma_f32_16x16x4_f32(false, a, false, bC, (short)0, c2, false, false);
        c3 = __builtin_amdgcn_wmma_f32_16x16x4_f32(false, a, false, bD, (short)0, c3, false, false);
    }

    float* crow = C + (size_t)(i0 + half * 8) * OUT_DIM + jb0 + lid;
    #pragma unroll
    for (int r = 0; r < 8; ++r) {
        crow[(size_t)r * OUT_DIM +  0] = c0[r];
        crow[(size_t)r * OUT_DIM + 16] = c1[r];
        crow[(size_t)r * OUT_DIM + 32] = c2[r];
        crow[(size_t)r * OUT_DIM + 48] = c3[r];
    }
}

// ---------------------------------------------------------------------------
extern "C" void kernel_launch(void* const* d_in, const int* in_sizes, int n_in,
                              void* d_out, int out_size, void* d_ws, size_t ws_size,
                              hipStream_t stream) {
    const float* input = (const float*)d_in[0];   // [65536, 256] f32
    const float* scale = (const float*)d_in[1];   // [2048] f32
    const int*   M_in  = (const int*)d_in[2];     // [2048]
    const int*   M     = (const int*)d_in[3];     // [2048]
    float*       out   = (float*)d_out;           // [65536, 256] f32
    float*       Wt    = (float*)d_ws;            // 256*256 f32 = 256 KB scratch

    zero_w_kernel<<<(K_DIM * OUT_DIM) / 256, 256, 0, stream>>>(Wt);
    scatter_w_kernel<<<NNZ_CNT / 256, 256, 0, stream>>>(scale, M_in, M, Wt);

    // 4096 row-tiles * 4 col-groups = 16384 waves / 8 waves-per-block
    gemm_wmma_kernel<<<(N_ROWS / 16) * 4 / 8, 256, 0, stream>>>(input, Wt, out);
}